// GraphConvolution_10428180595104
// MI455X (gfx1250) — compile-verified
//
#include <hip/hip_runtime.h>
#include <stdint.h>

// ---------------------------------------------------------------------------
// GraphConvolution (PPNP-style iterative solver) for MI455X / gfx1250.
// N=100000 rows, D=128 features, E=1.6M edges, 8 steps.
//  - Dense N x 128 x 128 GEMMs on the WMMA pipe in full FP32
//    (V_WMMA_F32_16X16X4_F32, K=128 as 32 chunks of 4).
//  - The four 128x128 B matrices (P1,P2,S1,S2) are pre-packed fragment-major
//    and staged into 256KB of LDS per workgroup via the gfx1250 async
//    global->LDS copy path (ASYNCcnt), then read with ds_load_b128.
//  - SpMM is edge-parallel with f32 L2 atomics (the 51MB Y matrix and both
//    edge lists are L2-resident on the 192MB L2).
// ---------------------------------------------------------------------------

typedef __attribute__((ext_vector_type(2))) float v2f;
typedef __attribute__((ext_vector_type(8))) float v8f;

#define DFEAT 128
#define ALPHA_C (1.0f / 3.0f)  // 1/(1 + 0 + lam0 + lam1), lam0=lam1=1

// Fragment-major packing of a 128x128 B matrix for V_WMMA_F32_16X16X4_F32:
//   float4 index = ct*512 + kcp*32 + lane   (ct: 8 col tiles, kcp: 16 double-K
//   chunks, lane: 32)
//   q-th float  = B[(8*kcp + 4*(q>>1) + 2*(lane>>4) + (q&1)) * 128
//                   + ct*16 + (lane&15)]
// One ds_load_b128 per lane per kcp feeds two K=4 WMMAs of that matrix.

// ---------------- prep: build packed P = H@H^T and S = H+H^T ---------------
// mat 0: P(H1)  mat 1: P(H2)  mat 2: S(H1)  mat 3: S(H2)
__global__ void prep_pack(const float* __restrict__ H1,
                          const float* __restrict__ H2,
                          float4* __restrict__ Bpack4) {
  int g = blockIdx.x * blockDim.x + threadIdx.x;  // 0 .. 4*4096-1
  if (g >= 4 * 4096) return;
  const int mat = g >> 12;
  const int r = g & 4095;
  const int ct = r >> 9;
  const int kcp = (r >> 5) & 15;
  const int lane = r & 31;
  const int kh = lane >> 4;
  const int col = ct * 16 + (lane & 15);
  const float* __restrict__ H = (mat & 1) ? H2 : H1;
  float v[4];
#pragma unroll
  for (int q = 0; q < 4; ++q) {
    const int kk = 8 * kcp + 4 * (q >> 1) + 2 * kh + (q & 1);
    if (mat < 2) {  // P = H @ H^T
      float acc = 0.0f;
#pragma unroll 8
      for (int t = 0; t < DFEAT; ++t)
        acc += H[kk * DFEAT + t] * H[col * DFEAT + t];
      v[q] = acc;
    } else {  // S = H + H^T
      v[q] = H[kk * DFEAT + col] + H[col * DFEAT + kk];
    }
  }
  Bpack4[g] = make_float4(v[0], v[1], v[2], v[3]);
}

// ---------------- zero workspace (float4 stores) ---------------------------
__global__ void zero_f4(float4* __restrict__ p, long n4) {
  long i = (long)blockIdx.x * blockDim.x + threadIdx.x;
  if (i < n4) p[i] = make_float4(0.f, 0.f, 0.f, 0.f);
}

// ---------------- SpMM: out[dst] += w * Y[src]  (edge per wave) ------------
__global__ void spmm_atomic(const float* __restrict__ Y,
                            const float* __restrict__ w,
                            const int* __restrict__ src,
                            const int* __restrict__ dst,
                            float* __restrict__ out, int E) {
  int gtid = blockIdx.x * blockDim.x + threadIdx.x;
  int lane = threadIdx.x & 31;
  int wid = gtid >> 5;
  int nwaves = (gridDim.x * blockDim.x) >> 5;
  for (int e = wid; e < E; e += nwaves) {
    int s = src[e];
    int d = dst[e];
    float we = w[e];
    const float4 yv = *(const float4*)(Y + (long)s * DFEAT + lane * 4);
    float* o = out + (long)d * DFEAT + lane * 4;
    atomicAdd(o + 0, we * yv.x);
    atomicAdd(o + 1, we * yv.y);
    atomicAdd(o + 2, we * yv.z);
    atomicAdd(o + 3, we * yv.w);
  }
}

// ---------------- fused WMMA update ---------------------------------------
// Per wave: 16-row slab. G1=Y@P1, G2=Y@P2, G3=AbY@S1, G4=AgY@S2 with
// V_WMMA_F32_16X16X4_F32, then epilogue:
//   yhat = (G3 - db*G1) + X0 + (dg*Y - AgY + G4 - dg*G2)
//   Ynew = (1-a)*Y + a*yhat/(db+dg+1)
// In-place safe: the full 16x128 Y A-fragment tile is register-cached before
// any write; epilogue of col-tile ct only touches cols [16ct,16ct+16).
__global__ __launch_bounds__(256) void fused_update(
    float* __restrict__ Y, const float* __restrict__ X0,
    const float* __restrict__ AbY, const float* __restrict__ AgY,
    const float4* __restrict__ Bpack4, const float* __restrict__ dbeta,
    const float* __restrict__ dgamma, int Nrows) {
  __shared__ float4 ldsB[4 * 4096];  // 256 KB: P1 | P2 | S1 | S2, packed

  // ---- stage packed B matrices into LDS via async global->LDS copies ----
  const int tid = threadIdx.x;
#pragma unroll 4
  for (int i = 0; i < 64; ++i) {
    const int idx = i * 256 + tid;
    // Low 32 bits of the flat shared-aperture address are the LDS byte
    // offset (ISA 10.2).
    unsigned laddr = (unsigned)(uintptr_t)&ldsB[idx];
    const float4* gp = Bpack4 + idx;
    asm volatile("global_load_async_to_lds_b128 %0, %1, off" ::"v"(laddr),
                 "v"(gp)
                 : "memory");
  }
  asm volatile("s_wait_asynccnt 0x0" ::: "memory");
  __syncthreads();

  const int lane = threadIdx.x & 31;
  const int wave = threadIdx.x >> 5;
  const int row0 = (blockIdx.x * 8 + wave) * 16;
  if (row0 >= Nrows) return;  // uniform per wave; EXEC stays all-ones

  const int m = lane & 15;   // A-matrix row within 16-row tile
  const int kh = lane >> 4;  // K-half selector (ISA 16x4 f32 A layout)
  const int n = lane & 15;   // C/D column within tile

  // Preload the full 16x128 Y row tile as A fragments (before any write).
  v2f yfrag[32];
  const float* yrow = Y + (long)(row0 + m) * DFEAT + kh * 2;
#pragma unroll
  for (int kc = 0; kc < 32; ++kc) yfrag[kc] = *(const v2f*)(yrow + kc * 4);

  const float* brow = AbY + (long)(row0 + m) * DFEAT + kh * 2;
  const float* grow = AgY + (long)(row0 + m) * DFEAT + kh * 2;

  for (int ct = 0; ct < 8; ++ct) {  // 8 column tiles of 16
    v8f g1 = {}, g2 = {}, g3 = {}, g4 = {};
    const int cb = ct * 16 + n;
    const int fb = ct * 512 + lane;  // float4 index base within a matrix
#pragma unroll
    for (int kcp = 0; kcp < 16; ++kcp) {  // 2 K-chunks per iteration
      const int fo = fb + kcp * 32;
      const float4 b1 = ldsB[0 * 4096 + fo];
      const float4 b2 = ldsB[1 * 4096 + fo];
      const float4 b3 = ldsB[2 * 4096 + fo];
      const float4 b4 = ldsB[3 * 4096 + fo];
      const v2f ab0 = *(const v2f*)(brow + 8 * kcp);
      const v2f ab1 = *(const v2f*)(brow + 8 * kcp + 4);
      const v2f ag0 = *(const v2f*)(grow + 8 * kcp);
      const v2f ag1 = *(const v2f*)(grow + 8 * kcp + 4);
      const v2f ya = yfrag[2 * kcp];
      const v2f yb = yfrag[2 * kcp + 1];
      v2f bl, bh;
      bl = (v2f){b1.x, b1.y};
      bh = (v2f){b1.z, b1.w};
      g1 = __builtin_amdgcn_wmma_f32_16x16x4_f32(false, ya, false, bl, (short)0,
                                                 g1, false, false);
      g1 = __builtin_amdgcn_wmma_f32_16x16x4_f32(false, yb, false, bh, (short)0,
                                                 g1, false, false);
      bl = (v2f){b2.x, b2.y};
      bh = (v2f){b2.z, b2.w};
      g2 = __builtin_amdgcn_wmma_f32_16x16x4_f32(false, ya, false, bl, (short)0,
                                                 g2, false, false);
      g2 = __builtin_amdgcn_wmma_f32_16x16x4_f32(false, yb, false, bh, (short)0,
                                                 g2, false, false);
      bl = (v2f){b3.x, b3.y};
      bh = (v2f){b3.z, b3.w};
      g3 = __builtin_amdgcn_wmma_f32_16x16x4_f32(false, ab0, false, bl,
                                                 (short)0, g3, false, false);
      g3 = __builtin_amdgcn_wmma_f32_16x16x4_f32(false, ab1, false, bh,
                                                 (short)0, g3, false, false);
      bl = (v2f){b4.x, b4.y};
      bh = (v2f){b4.z, b4.w};
      g4 = __builtin_amdgcn_wmma_f32_16x16x4_f32(false, ag0, false, bl,
                                                 (short)0, g4, false, false);
      g4 = __builtin_amdgcn_wmma_f32_16x16x4_f32(false, ag1, false, bh,
                                                 (short)0, g4, false, false);
    }
    // Epilogue: C/D layout — lane l, vgpr r holds (M = r + 8*(l>>4), N = l&15)
#pragma unroll
    for (int r = 0; r < 8; ++r) {
      const int row = row0 + r + 8 * kh;
      const long off = (long)row * DFEAT + cb;
      const float yv = Y[off];
      const float x0 = X0[off];
      const float agv = AgY[off];
      const float db = dbeta[row];
      const float dg = dgamma[row];
      const float q = db + dg + 1.0f;
      const float yhat =
          (g3[r] - db * g1[r]) + x0 + (dg * yv - agv + g4[r] - dg * g2[r]);
      Y[off] = (1.0f - ALPHA_C) * yv + ALPHA_C * (yhat / q);
    }
  }
}

// ---------------------------------------------------------------------------
extern "C" void kernel_launch(void* const* d_in, const int* in_sizes, int n_in,
                              void* d_out, int out_size, void* d_ws,
                              size_t ws_size, hipStream_t stream) {
  const float* X = (const float*)d_in[0];
  const float* H1 = (const float*)d_in[1];
  const float* H2 = (const float*)d_in[2];
  const float* w_beta = (const float*)d_in[3];
  const float* w_gamma = (const float*)d_in[4];
  const float* d_beta = (const float*)d_in[5];
  const float* d_gamma = (const float*)d_in[6];
  const int* src_beta = (const int*)d_in[7];
  const int* dst_beta = (const int*)d_in[8];
  const int* src_gamma = (const int*)d_in[9];
  const int* dst_gamma = (const int*)d_in[10];

  const int N = in_sizes[5];  // d_beta has N elements
  const int E = in_sizes[7];  // src_beta has E elements

  float* Y = (float*)d_out;
  float* ws = (float*)d_ws;
  float* AbY = ws;                        // N*D floats
  float* AgY = AbY + (size_t)N * DFEAT;   // N*D floats
  float4* Bpack4 = (float4*)(AgY + (size_t)N * DFEAT);  // 4*4096 float4

  // One-time per call: packed S/P matrices + Y <- X
  prep_pack<<<(4 * 4096 + 255) / 256, 256, 0, stream>>>(H1, H2, Bpack4);
  hipMemcpyAsync(Y, X, (size_t)N * DFEAT * sizeof(float),
                 hipMemcpyDeviceToDevice, stream);

  const long n4 = (long)N * DFEAT * 2 / 4;  // AbY+AgY are contiguous
  const int zgrid = (int)((n4 + 255) / 256);
  const int ggrid = (N + 127) / 128;  // 8 waves/block * 16 rows/wave

  for (int step = 0; step < 8; ++step) {
    zero_f4<<<zgrid, 256, 0, stream>>>((float4*)AbY, n4);
    spmm_atomic<<<4096, 256, 0, stream>>>(Y, w_beta, src_beta, dst_beta, AbY,
                                          E);
    spmm_atomic<<<4096, 256, 0, stream>>>(Y, w_gamma, src_gamma, dst_gamma,
                                          AgY, E);
    fused_update<<<ggrid, 256, 0, stream>>>(Y, X, AbY, AgY, Bpack4, d_beta,
                                            d_gamma, N);
  }
}